// InstanceContrastiveLoss_79121887527567
// MI455X (gfx1250) — compile-verified
//
#include <hip/hip_runtime.h>
#include <hip/hip_bf16.h>
#include <math.h>

#define TOT      262144      // H*W
#define NUM_IDS  64
#define EPSF     1e-6f
#define MARGINF  1.0f

typedef __attribute__((ext_vector_type(2))) float v2f;
typedef __attribute__((ext_vector_type(8))) float v8f;

// ---------------------------------------------------------------------------
// Threefry-2x32 (JAX default PRNG), 20 rounds
// ---------------------------------------------------------------------------
__device__ __forceinline__ void threefry2x32(unsigned k0, unsigned k1,
                                             unsigned x0, unsigned x1,
                                             unsigned &o0, unsigned &o1) {
  unsigned ks[3] = {k0, k1, k0 ^ k1 ^ 0x1BD11BDAu};
  const int R[8] = {13, 15, 26, 6, 17, 29, 16, 24};
  x0 += ks[0];
  x1 += ks[1];
#pragma unroll
  for (int g = 0; g < 5; ++g) {
#pragma unroll
    for (int j = 0; j < 4; ++j) {
      int r = R[(g & 1) * 4 + j];
      x0 += x1;
      x1 = (x1 << r) | (x1 >> (32 - r));
      x1 ^= x0;
    }
    x0 += ks[(g + 1) % 3];
    x1 += ks[(g + 2) % 3] + (unsigned)(g + 1);
  }
  o0 = x0;
  o1 = x1;
}

// ---------------------------------------------------------------------------
// Kernel 0: zero scratch accumulators
// ---------------------------------------------------------------------------
__global__ void icl_init_kernel(int* __restrict__ cnt, float* __restrict__ acc) {
  int t = threadIdx.x;
  if (t < NUM_IDS) cnt[t] = 0;
  if (t == NUM_IDS) { acc[0] = 0.f; acc[1] = 0.f; }
}

// ---------------------------------------------------------------------------
// Kernel 1: 64-bin histogram of the instance mask (per-id pixel counts)
// 256 blocks x 256 threads, 1024 pixels/block
// ---------------------------------------------------------------------------
__global__ void icl_count_kernel(const int* __restrict__ mask, int* __restrict__ cnt) {
  __shared__ int h[NUM_IDS];
  const int tid = threadIdx.x;
  if (tid < NUM_IDS) h[tid] = 0;
  __syncthreads();
  const int base = blockIdx.x * 1024;
#pragma unroll
  for (int j = 0; j < 4; ++j) {
    int v = mask[base + tid + j * 256] & 63;
    atomicAdd(&h[v], 1);
  }
  __syncthreads();
  if (tid < NUM_IDS) atomicAdd(&cnt[tid], h[tid]);
}

// ---------------------------------------------------------------------------
// Kernel 2: per-instance resolve — first/second positive pixel and the
// (r+1)-th negative pixel, r from JAX threefry randint. One block / instance.
// ---------------------------------------------------------------------------
__global__ void icl_resolve_kernel(const int* __restrict__ mask,
                                   const int* __restrict__ cnt,
                                   int* __restrict__ firstO,
                                   int* __restrict__ secondO,
                                   int* __restrict__ nidxO,
                                   int* __restrict__ validO) {
  const int inst = (int)blockIdx.x + 1;  // 1..63
  const int tid = threadIdx.x;
  __shared__ int s_pc[256], s_p0[256], s_p1[256];
  __shared__ int s_done;

  int c = 0, ncnt = 0;
  unsigned target = 0;
  if (tid == 0) {
    c = cnt[inst];
    ncnt = TOT - c;
    // key = threefry_seed(42) = (0,42); key' = fold_in(key, inst)
    unsigned fk0, fk1;
    threefry2x32(0u, 42u, 0u, (unsigned)inst, fk0, fk1);
    // split(key',2): counts iota(4) -> pairs (0,2),(1,3); k1=(A0,B0), k2=(A1,B1)
    unsigned a0, a1, b0, b1;
    threefry2x32(fk0, fk1, 0u, 2u, a0, a1);
    threefry2x32(fk0, fk1, 1u, 3u, b0, b1);
    // random_bits(k,32,()) = first word of threefry(k,(0,0))
    unsigned h0, h1, l0, l1;
    threefry2x32(a0, b0, 0u, 0u, h0, h1);
    threefry2x32(a1, b1, 0u, 0u, l0, l1);
    unsigned span = (unsigned)((ncnt > 1) ? ncnt : 1);
    unsigned m = 65536u % span;     // all uint32 wrapping, as in jax _randint
    m = (m * m) % span;
    unsigned r = (h0 * m + l0) % span;
    target = r + 1u;                // need (r+1)-th negative
  }

  int posSeen = 0, firstIdx = 0, secondIdx = 0, nfound = 0, nidx = 0;
  unsigned negRun = 0;

  for (int tile = 0; tile < 32; ++tile) {      // 32 tiles x 8192 pixels
    const int base = tile * 8192 + tid * 32;   // contiguous 32-pixel segment
    int pc = 0, p0 = 0, p1 = 0;
#pragma unroll
    for (int j = 0; j < 32; ++j) {
      if (mask[base + j] == inst) {
        if (pc == 0) p0 = base + j;
        else if (pc == 1) p1 = base + j;
        pc++;
      }
    }
    s_pc[tid] = pc; s_p0[tid] = p0; s_p1[tid] = p1;
    __syncthreads();
    if (tid == 0) {
      for (int t = 0; t < 256; ++t) {
        const int tpc = s_pc[t];
        if (posSeen < 2 && tpc > 0) {
          if (posSeen == 0) {
            firstIdx = s_p0[t]; posSeen = 1;
            if (tpc >= 2) { secondIdx = s_p1[t]; posSeen = 2; }
          } else {
            secondIdx = s_p0[t]; posSeen = 2;
          }
        }
        const unsigned nc = (unsigned)(32 - tpc);
        if (!nfound && negRun + nc >= target) {
          const int b2 = tile * 8192 + t * 32;
          unsigned run = negRun;
          for (int j = 0; j < 32; ++j) {
            if (mask[b2 + j] != inst) {
              run++;
              if (run == target) { nidx = b2 + j; nfound = 1; break; }
            }
          }
        }
        negRun += nc;
      }
      s_done = (posSeen >= 2 && nfound) ? 1 : 0;
    }
    __syncthreads();
    if (s_done) break;
  }

  if (tid == 0) {
    firstO[inst] = firstIdx;
    secondO[inst] = secondIdx;
    nidxO[inst] = nidx;
    validO[inst] = (c >= 2 && ncnt > 0) ? 1 : 0;
  }
}

// ---------------------------------------------------------------------------
// Kernel 3: triplet loss via V_WMMA_F32_16X16X4_F32.
// 8 blocks x 256 threads; each block handles 8 instances (16 slots:
// slot 2q = a-p+eps, slot 2q+1 = a-n+eps). D-diagonal = squared norms.
// ---------------------------------------------------------------------------
__global__ void icl_loss_kernel(const float* __restrict__ feats,
                                const int* __restrict__ firstA,
                                const int* __restrict__ secondA,
                                const int* __restrict__ nidxA,
                                const int* __restrict__ validA,
                                float* __restrict__ acc) {
  __shared__ float av[8][512];
  __shared__ float pv[8][512];
  __shared__ float nv[8][512];
  __shared__ float sq[16];
  __shared__ int s_first[8], s_second[8], s_nidx[8], s_valid[8];
  const int tid = threadIdx.x;

  if (tid < 8) {
    int q = blockIdx.x * 8 + tid;   // 0..63, q==63 unused
    int f = 0, s = 0, n = 0, v = 0;
    if (q < 63) {
      int inst = q + 1;
      f = firstA[inst]; s = secondA[inst]; n = nidxA[inst]; v = validA[inst];
      f = (f >= 0 && f < TOT) ? f : 0;
      s = (s >= 0 && s < TOT) ? s : 0;
      n = (n >= 0 && n < TOT) ? n : 0;
    }
    s_first[tid] = f; s_second[tid] = s; s_nidx[tid] = n; s_valid[tid] = v;
  }
  __syncthreads();

  // Stage the 8x3 feature columns (512 dims each, row stride = TOT) into LDS.
  for (int e = tid; e < 8 * 3 * 512; e += 256) {
    const int vec = e >> 9;            // 0..23
    const int k = e & 511;             // feature dim
    const int s = vec / 3;
    const int w = vec - 3 * s;         // 0=anchor,1=pos,2=neg
    const int pix = (w == 0) ? s_first[s] : (w == 1) ? s_second[s] : s_nidx[s];
    const float val = feats[(size_t)k * (size_t)TOT + (size_t)pix];
    if (w == 0) av[s][k] = val;
    else if (w == 1) pv[s][k] = val;
    else nv[s][k] = val;
  }
  __syncthreads();

  // Wave 0 only: EXEC all-ones, 128 x V_WMMA_F32_16X16X4_F32, K = 0..511.
  // f32 16x4 A and 4x16 B share the same per-lane packing: lane l holds
  // slot (l%16)'s K-pair selected by lane-half -> one float2 feeds A and B.
  if (tid < 32) {
    const int slot = tid & 15;
    const int q = slot >> 1;
    const int which = slot & 1;
    const int khalf = tid >> 4;        // 0: K={4t,4t+1}, 1: K={4t+2,4t+3}
    const float* aV = av[q];
    const float* xV = which ? nv[q] : pv[q];
    v8f c = {0.f, 0.f, 0.f, 0.f, 0.f, 0.f, 0.f, 0.f};
#pragma unroll 4
    for (int t = 0; t < 128; ++t) {
      const int k = 4 * t + 2 * khalf;
      v2f ab;
      ab.x = aV[k]     - xV[k]     + EPSF;
      ab.y = aV[k + 1] - xV[k + 1] + EPSF;
      // 8 args: (neg_a, A, neg_b, B, c_mod, C, reuse_a, reuse_b)
      c = __builtin_amdgcn_wmma_f32_16x16x4_f32(false, ab, false, ab,
                                                (short)0, c, false, false);
    }
    // Extract diagonal D[i][i]: i<8 -> lane i, VGPR i; i>=8 -> lane i+16, VGPR i-8
    if (tid < 8)        sq[tid]      = c[tid];
    else if (tid >= 24) sq[tid - 16] = c[tid - 24];
  }
  __syncthreads();

  if (tid == 0) {
    float lsum = 0.f, lcnt = 0.f;
#pragma unroll
    for (int s = 0; s < 8; ++s) {
      if (s_valid[s]) {
        const float dp = sqrtf(sq[2 * s]);
        const float dn = sqrtf(sq[2 * s + 1]);
        lsum += fmaxf(dp - dn + MARGINF, 0.f);
        lcnt += 1.f;
      }
    }
    atomicAdd(&acc[0], lsum);
    atomicAdd(&acc[1], lcnt);
  }
}

// ---------------------------------------------------------------------------
// Kernel 4: finalize mean
// ---------------------------------------------------------------------------
__global__ void icl_finalize_kernel(const float* __restrict__ acc,
                                    float* __restrict__ out) {
  if (threadIdx.x == 0) {
    const float s = acc[0], c = acc[1];
    out[0] = (c > 0.f) ? (s / fmaxf(c, 1.f)) : 0.f;
  }
}

extern "C" void kernel_launch(void* const* d_in, const int* in_sizes, int n_in,
                              void* d_out, int out_size, void* d_ws, size_t ws_size,
                              hipStream_t stream) {
  const float* feats = (const float*)d_in[0];   // (2,256,512,512) f32 -> (512, 262144)
  const int* mask = (const int*)d_in[1];        // (512,512) i32

  int* cnt     = (int*)d_ws;          // 64
  int* firstA  = cnt + 64;            // 64
  int* secondA = firstA + 64;         // 64
  int* nidxA   = secondA + 64;        // 64
  int* validA  = nidxA + 64;          // 64
  float* acc   = (float*)(validA + 64); // 2 floats
  float* out   = (float*)d_out;

  icl_init_kernel<<<1, 128, 0, stream>>>(cnt, acc);
  icl_count_kernel<<<256, 256, 0, stream>>>(mask, cnt);
  icl_resolve_kernel<<<63, 256, 0, stream>>>(mask, cnt, firstA, secondA, nidxA, validA);
  icl_loss_kernel<<<8, 256, 0, stream>>>(feats, firstA, secondA, nidxA, validA, acc);
  icl_finalize_kernel<<<1, 32, 0, stream>>>(acc, out);
}